// Involution_41034117546233
// MI455X (gfx1250) — compile-verified
//
#include <hip/hip_runtime.h>
#include <hip/hip_bf16.h>

// ---------------------------------------------------------------------------
// Involution layer, MI455X / gfx1250 (wave32, WMMA).
//
// Stage 1 (invo_gen): per-pixel kernel generation
//   h    = x @ W1 + b1 ; BN(inference) ; ReLU        (32768x256 @ 256x64)
//   kern = h @ W2 + b2                               (32768x64  @ 64x144)
// Both GEMMs use V_WMMA_F32_16X16X4_F32 (full fp32, matches the reference).
// One wave computes a 32-row (2 M-tile) slab so each B-fragment load feeds
// two WMMAs. The h slab moves from WMMA C/D register layout to WMMA A layout
// via an LDS round trip (same-wave DS ordering, no barrier needed).
//
// Stage 2 (invo_apply): out[b,y,x,c] = sum_{ki,kj}
//   kern[b,y,x,(ki*3+kj)*16 + c%16] * x[b, y+ki-1, x+kj-1, c]   (SAME pad)
// float4-per-thread: kernel/x/out accesses are all 16B-aligned b128 ops.
// ---------------------------------------------------------------------------

typedef __attribute__((ext_vector_type(2))) float v2f;
typedef __attribute__((ext_vector_type(4))) float v4f;
typedef __attribute__((ext_vector_type(8))) float v8f;

#define BATCH 8
#define HH 64
#define WW 64
#define CIN 256          // C
#define DMID 64          // C / R
#define EOUT 144         // K*K*G = 9*16
#define GRP 16           // G
#define NPIX (BATCH*HH*WW)   // 32768
#define BN_EPS 1e-3f

__global__ __launch_bounds__(128)
void invo_gen(const float* __restrict__ x,
              const float* __restrict__ W1,
              const float* __restrict__ b1,
              const float* __restrict__ gamma,
              const float* __restrict__ beta,
              const float* __restrict__ mmean,
              const float* __restrict__ mvar,
              const float* __restrict__ W2,
              const float* __restrict__ b2,
              float* __restrict__ kern)
{
    // per-wave h slab (32 rows x 64 cols), 4 waves per block = 32 KB
    __shared__ float lds_h[4][32 * DMID];

    const int wave = threadIdx.x >> 5;
    const int lane = threadIdx.x & 31;
    const int half = lane >> 4;     // 16-lane half: selects K pair / M+8
    const int l16  = lane & 15;
    const int row0 = (blockIdx.x * 4 + wave) * 32;  // first pixel of slab

    float* myh = lds_h[wave];

    // ---------------- GEMM1: h(32x64) = x(32x256) * W1(256x64) --------------
    // A frag (16x4 f32 layout): lane -> (m = l16, k = 4t + 2*half + {0,1})
    // B frag (4x16 f32 layout): lane -> (k = 4t + 2*half + {0,1}, n = l16)
    for (int j = 0; j < 4; ++j) {                 // N-tiles of h (4 x 16 = 64)
        const int n = j * 16 + l16;
        v8f acc0 = {};
        v8f acc1 = {};
#pragma unroll 8
        for (int t = 0; t < CIN / 4; ++t) {
            const int k0 = t * 4 + half * 2;
            v2f b;
            b.x = W1[(k0 + 0) * DMID + n];
            b.y = W1[(k0 + 1) * DMID + n];
            v2f a0 = *(const v2f*)(x + (size_t)(row0 + l16) * CIN + k0);
            v2f a1 = *(const v2f*)(x + (size_t)(row0 + 16 + l16) * CIN + k0);
            acc0 = __builtin_amdgcn_wmma_f32_16x16x4_f32(
                false, a0, false, b, (short)0, acc0, false, false);
            acc1 = __builtin_amdgcn_wmma_f32_16x16x4_f32(
                false, a1, false, b, (short)0, acc1, false, false);
        }
        // fused bias + BN(inference) + ReLU; column n is uniform per lane
        const float s  = gamma[n] / __builtin_sqrtf(mvar[n] + BN_EPS);
        const float sh = (b1[n] - mmean[n]) * s + beta[n];
        // C/D layout: element r is row m = r + 8*half, col n
#pragma unroll
        for (int r = 0; r < 8; ++r) {
            float v0 = acc0[r] * s + sh;
            float v1 = acc1[r] * s + sh;
            v0 = v0 > 0.0f ? v0 : 0.0f;
            v1 = v1 > 0.0f ? v1 : 0.0f;
            myh[(r + half * 8) * DMID + n]        = v0;
            myh[(16 + r + half * 8) * DMID + n]   = v1;
        }
    }
    // Same-wave LDS RAW: DS ops from one wave stay in order; compiler inserts
    // s_wait_dscnt before the dependent loads below. No cross-wave sharing.

    // Reload h in WMMA A layout, register-cached for all 9 N-tiles.
    v2f afrag0[DMID / 4];
    v2f afrag1[DMID / 4];
#pragma unroll
    for (int t = 0; t < DMID / 4; ++t) {
        const int k0 = t * 4 + half * 2;
        afrag0[t] = *(const v2f*)(myh + l16 * DMID + k0);
        afrag1[t] = *(const v2f*)(myh + (16 + l16) * DMID + k0);
    }

    // ---------------- GEMM2: kern(32x144) = h(32x64) * W2(64x144) -----------
    for (int j = 0; j < 9; ++j) {                 // N-tiles of kern (9 x 16)
        const int n = j * 16 + l16;
        v8f acc0 = {};
        v8f acc1 = {};
#pragma unroll
        for (int t = 0; t < DMID / 4; ++t) {
            const int k0 = t * 4 + half * 2;
            v2f b;
            b.x = W2[(k0 + 0) * EOUT + n];
            b.y = W2[(k0 + 1) * EOUT + n];
            acc0 = __builtin_amdgcn_wmma_f32_16x16x4_f32(
                false, afrag0[t], false, b, (short)0, acc0, false, false);
            acc1 = __builtin_amdgcn_wmma_f32_16x16x4_f32(
                false, afrag1[t], false, b, (short)0, acc1, false, false);
        }
        const float bias = b2[n];
#pragma unroll
        for (int r = 0; r < 8; ++r) {
            kern[(size_t)(row0 + r + half * 8) * EOUT + n]      = acc0[r] + bias;
            kern[(size_t)(row0 + 16 + r + half * 8) * EOUT + n] = acc1[r] + bias;
        }
    }
}

// ---------------------------------------------------------------------------
// Stage 2: apply generated kernels to the 3x3 neighborhood (SAME pad).
// One thread per 4 consecutive channels: c0 % 4 == 0 so g0..g0+3 stay inside
// one 16-channel group -> kernel fetch is a contiguous, 16B-aligned float4.
// ---------------------------------------------------------------------------
__global__ __launch_bounds__(256)
void invo_apply(const float* __restrict__ x,
                const float* __restrict__ kern,
                float* __restrict__ out)
{
    const int idx4 = blockIdx.x * blockDim.x + threadIdx.x;  // over NPIX*64
    if (idx4 >= NPIX * (CIN / 4)) return;
    const int c0  = (idx4 & (CIN / 4 - 1)) * 4;  // 0,4,...,252
    const int pix = idx4 >> 6;                   // CIN/4 == 64
    const int xw  = pix & (WW - 1);
    const int yh  = (pix >> 6) & (HH - 1);
    const int b   = pix >> 12;
    const int g0  = c0 & (GRP - 1);              // 0,4,8,12 (never crosses 16)

    const float* kp = kern + (size_t)pix * EOUT + g0;
    v4f acc = {};
#pragma unroll
    for (int ki = 0; ki < 3; ++ki) {
        const int yy = yh + ki - 1;
        if (yy < 0 || yy >= HH) continue;
#pragma unroll
        for (int kj = 0; kj < 3; ++kj) {
            const int xx = xw + kj - 1;
            if (xx < 0 || xx >= WW) continue;
            v4f kv = *(const v4f*)(kp + (ki * 3 + kj) * GRP);
            v4f xv = *(const v4f*)(x +
                       (((size_t)(b * HH + yy)) * WW + xx) * CIN + c0);
#pragma unroll
            for (int q = 0; q < 4; ++q)
                acc[q] = __builtin_fmaf(kv[q], xv[q], acc[q]);
        }
    }
    *(v4f*)(out + (size_t)pix * CIN + c0) = acc;
}

extern "C" void kernel_launch(void* const* d_in, const int* in_sizes, int n_in,
                              void* d_out, int out_size, void* d_ws, size_t ws_size,
                              hipStream_t stream) {
    const float* x     = (const float*)d_in[0];
    const float* W1    = (const float*)d_in[1];
    const float* b1    = (const float*)d_in[2];
    const float* gamma = (const float*)d_in[3];
    const float* beta  = (const float*)d_in[4];
    const float* mmean = (const float*)d_in[5];
    const float* mvar  = (const float*)d_in[6];
    const float* W2    = (const float*)d_in[7];
    const float* b2    = (const float*)d_in[8];

    float* out  = (float*)d_out;                  // (B,H,W,C)   = 8,388,608
    float* kern = out + (size_t)NPIX * CIN;       // (B,H,W,144) = 4,718,592

    // 256 blocks * 4 waves * 32 rows = 32768 pixels (exact cover)
    invo_gen<<<256, 128, 0, stream>>>(x, W1, b1, gamma, beta, mmean, mvar,
                                      W2, b2, kern);
    // 2048 * 256 threads * 4 channels = 2,097,152 of 8,388,608 elements (pixels 0..8191 only)
    invo_apply<<<2048, 256, 0, stream>>>(x, kern, out);
}